// GraphSAGE_37580963840349
// MI455X (gfx1250) — compile-verified
//
#include <hip/hip_runtime.h>
#include <hip/hip_bf16.h>

typedef float v2f __attribute__((ext_vector_type(2)));
typedef float v8f __attribute__((ext_vector_type(8)));

// ---------------------------------------------------------------- utilities
__global__ void zero4_kernel(float4* __restrict__ p, int n4) {
  int i = blockIdx.x * blockDim.x + threadIdx.x;
  if (i < n4) p[i] = make_float4(0.f, 0.f, 0.f, 0.f);
}

// ------------------------------------------- layer 1: F=1 scatter + degrees
__global__ void scatter1_kernel(const float* __restrict__ x,
                                const int* __restrict__ src,
                                const int* __restrict__ dst,
                                float* __restrict__ s1,     // [N*4] sums
                                float* __restrict__ deg,    // [N]
                                int E) {
  int e = blockIdx.x * blockDim.x + threadIdx.x;
  if (e >= E) return;
  int sn = src[e], dn = dst[e];
  const float4 v = *(const float4*)(x + (size_t)sn * 4);
  float* p = s1 + (size_t)dn * 4;
  atomicAdd(p + 0, v.x); atomicAdd(p + 1, v.y);
  atomicAdd(p + 2, v.z); atomicAdd(p + 3, v.w);
  atomicAdd(deg + dn, 1.0f);
}

__global__ void invdeg_kernel(const float* __restrict__ deg,
                              float* __restrict__ inv, int N) {
  int n = blockIdx.x * blockDim.x + threadIdx.x;
  if (n < N) inv[n] = 1.0f / fmaxf(deg[n], 1.0f);
}

// layer 1 GEMM is an outer product (F=1): h[g*64+c] = relu(agg*W1l[c]+b+x*W1r[c])
__global__ void sage1_kernel(const float* __restrict__ s1,
                             const float* __restrict__ inv,
                             const float* __restrict__ x,
                             const float* __restrict__ W1l,
                             const float* __restrict__ b1l,
                             const float* __restrict__ W1r,
                             float* __restrict__ h, int total /* N*256 */) {
  int idx = blockIdx.x * blockDim.x + threadIdx.x;
  if (idx >= total) return;
  int g = idx >> 6;          // row = n*4 + m
  int c = idx & 63;
  float a = s1[g] * inv[g >> 2];
  float v = a * W1l[c] + b1l[c] + x[g] * W1r[c];
  h[idx] = v > 0.f ? v : 0.f;     // flat g*64+c == idx
}

// ------------------------------- layers 2/3: 256-float scatter-add per edge
__global__ void scatter_add_kernel(const float* __restrict__ h,
                                   const int* __restrict__ src,
                                   const int* __restrict__ dst,
                                   float* __restrict__ s, int total /* E*64 */) {
  int idx = blockIdx.x * blockDim.x + threadIdx.x;
  if (idx >= total) return;
  int e = idx >> 6, q = idx & 63;
  int sn = src[e], dn = dst[e];
  const float4 v = *(const float4*)(h + (size_t)sn * 256 + q * 4);
  float* p = s + (size_t)dn * 256 + q * 4;
  atomicAdd(p + 0, v.x); atomicAdd(p + 1, v.y);
  atomicAdd(p + 2, v.z); atomicAdd(p + 3, v.w);
}

// ----------------------- fused dual-GEMM + bias + ReLU using fp32 WMMA tiles
// out[g,c] = relu( (agg[g,:]*inv[g>>2]) @ Wl^T + bl + self[g,:] @ Wr^T )
// One wave owns 16 rows (4 nodes) x 64 cols; K=64 in 16 steps of 4.
// `out` may alias `agg` (rows are fully read before being written).
__global__ void sage_gemm_kernel(const float* __restrict__ agg,
                                 const float* __restrict__ self,
                                 const float* __restrict__ inv,
                                 const float* __restrict__ Wl,
                                 const float* __restrict__ bl,
                                 const float* __restrict__ Wr,
                                 float* __restrict__ out, int nrows /* 4N */) {
  __shared__ float sWl[64 * 64];   // stored transposed: sWl[k*64 + c] = Wl[c,k]
  __shared__ float sWr[64 * 64];
  __shared__ float sb[64];
  for (int i = threadIdx.x; i < 64 * 64; i += blockDim.x) {
    int k = i >> 6, c = i & 63;
    sWl[i] = Wl[c * 64 + k];
    sWr[i] = Wr[c * 64 + k];
  }
  if (threadIdx.x < 64) sb[threadIdx.x] = bl[threadIdx.x];
  __syncthreads();

  int wave = blockIdx.x * (blockDim.x >> 5) + (threadIdx.x >> 5);
  int lane = threadIdx.x & 31;
  int rowbase = wave * 16;
  if (rowbase >= nrows) return;          // wave-uniform exit (EXEC stays full)

  int mrow = lane & 15;                  // A-row / B-col index for this lane
  int koff = (lane >> 4) << 1;           // 0 or 2 (K pair select per half-wave)
  int grow = rowbase + mrow;
  float sc = inv[grow >> 2];

  v8f acc0 = {}, acc1 = {}, acc2 = {}, acc3 = {};
  v8f acc[4] = {acc0, acc1, acc2, acc3};

  for (int ks = 0; ks < 16; ++ks) {
    int kb = ks * 4;
    v2f a, xv;
    a.x  = agg[(size_t)grow * 64 + kb + koff]     * sc;
    a.y  = agg[(size_t)grow * 64 + kb + koff + 1] * sc;
    xv.x = self[(size_t)grow * 64 + kb + koff];
    xv.y = self[(size_t)grow * 64 + kb + koff + 1];
#pragma unroll
    for (int ct = 0; ct < 4; ++ct) {
      int cg = ct * 16 + mrow;
      v2f bl2, br2;
      bl2.x = sWl[(kb + koff) * 64 + cg];
      bl2.y = sWl[(kb + koff + 1) * 64 + cg];
      br2.x = sWr[(kb + koff) * 64 + cg];
      br2.y = sWr[(kb + koff + 1) * 64 + cg];
      acc[ct] = __builtin_amdgcn_wmma_f32_16x16x4_f32(
          false, a, false, bl2, (short)0, acc[ct], false, false);
      acc[ct] = __builtin_amdgcn_wmma_f32_16x16x4_f32(
          false, xv, false, br2, (short)0, acc[ct], false, false);
    }
  }

  // epilogue: C/D layout -> lane col = lane&15, row = v + 8*(lane>>4)
  int rhalf = (lane >> 4) * 8;
#pragma unroll
  for (int ct = 0; ct < 4; ++ct) {
    int cg = ct * 16 + mrow;
    float bias = sb[cg];
#pragma unroll
    for (int v = 0; v < 8; ++v) {
      int r = rowbase + rhalf + v;
      float val = acc[ct][v] + bias;
      out[(size_t)r * 64 + cg] = val > 0.f ? val : 0.f;
    }
  }
}

// ----------------------------------------------------- global mean pool (+count)
__global__ void pool_kernel(const float* __restrict__ h,
                            const int* __restrict__ batch,
                            float* __restrict__ pooled,   // [64*256] sums
                            float* __restrict__ gcnt,     // [64]
                            int total /* N*64 */) {
  int idx = blockIdx.x * blockDim.x + threadIdx.x;
  if (idx >= total) return;
  int n = idx >> 6, q = idx & 63;
  int b = batch[n];
  const float4 v = *(const float4*)(h + (size_t)n * 256 + q * 4);
  float* p = pooled + (size_t)b * 256 + q * 4;
  atomicAdd(p + 0, v.x); atomicAdd(p + 1, v.y);
  atomicAdd(p + 2, v.z); atomicAdd(p + 3, v.w);
  if (q == 0) atomicAdd(gcnt + b, 1.0f);
}

// ------------------------------------- final linear + sigmoid + mean over M
__global__ void final_kernel(const float* __restrict__ pooled,
                             const float* __restrict__ gcnt,
                             const float* __restrict__ Wlin,
                             const float* __restrict__ blin,
                             float* __restrict__ out, int B) {
  int b = threadIdx.x;
  if (b >= B) return;
  float inv = 1.0f / fmaxf(gcnt[b], 1.0f);
  float accum = 0.f;
  for (int j = 0; j < 4; ++j) {
    float z = blin[0];
    for (int c = 0; c < 64; ++c)
      z += pooled[(size_t)b * 256 + j * 64 + c] * inv * Wlin[c];
    accum += 1.0f / (1.0f + expf(-z));
  }
  out[b] = accum * 0.25f;
}

// ---------------------------------------------------------------- dispatcher
extern "C" void kernel_launch(void* const* d_in, const int* in_sizes, int n_in,
                              void* d_out, int out_size, void* d_ws, size_t ws_size,
                              hipStream_t stream) {
  const float* x    = (const float*)d_in[0];     // [N,1,4]
  const int*   ei   = (const int*)d_in[1];       // [2,E]
  const int*   batch= (const int*)d_in[2];       // [N]
  const float* W1l  = (const float*)d_in[3];
  const float* b1l  = (const float*)d_in[4];
  const float* W1r  = (const float*)d_in[5];
  const float* W2l  = (const float*)d_in[6];
  const float* b2l  = (const float*)d_in[7];
  const float* W2r  = (const float*)d_in[8];
  const float* W3l  = (const float*)d_in[9];
  const float* b3l  = (const float*)d_in[10];
  const float* W3r  = (const float*)d_in[11];
  const float* Wlin = (const float*)d_in[12];
  const float* blin = (const float*)d_in[13];
  float* out = (float*)d_out;

  const int N = in_sizes[0] / 4;     // 100000
  const int E = in_sizes[1] / 2;     // 1000000
  const int* src = ei;
  const int* dst = ei + E;
  const int nrows = N * 4;           // 400000 (multiple of 16)
  const int feat  = N * 256;

  char* w = (char*)d_ws;
  const size_t szH = (size_t)feat * sizeof(float);           // 102.4 MB
  float* hA     = (float*)w;
  float* hB     = (float*)(w + szH);
  float* s1     = (float*)(w + 2 * szH);                     // [N*4]
  float* deg    = s1 + (size_t)N * 4;                        // [N]
  float* inv    = deg + N;                                   // [N]
  float* pooled = inv + N;                                   // [64*256]
  float* gcnt   = pooled + 64 * 256;                         // [64]

  const int TB = 256;
  auto blocks = [](int n, int tb) { return (n + tb - 1) / tb; };

  // ---- layer 1 ----
  zero4_kernel<<<blocks((N * 5) / 4, TB), TB, 0, stream>>>((float4*)s1, (N * 5) / 4); // s1+deg
  scatter1_kernel<<<blocks(E, TB), TB, 0, stream>>>(x, src, dst, s1, deg, E);
  invdeg_kernel<<<blocks(N, TB), TB, 0, stream>>>(deg, inv, N);
  sage1_kernel<<<blocks(feat, TB), TB, 0, stream>>>(s1, inv, x, W1l, b1l, W1r, hA, feat);

  const int gemm_blocks = (nrows / 16 + (TB / 32) - 1) / (TB / 32);

  // ---- layer 2: scatter hA -> hB, GEMM(hB agg, hA self) -> hB ----
  zero4_kernel<<<blocks(feat / 4, TB), TB, 0, stream>>>((float4*)hB, feat / 4);
  scatter_add_kernel<<<blocks(E * 64, TB), TB, 0, stream>>>(hA, src, dst, hB, E * 64);
  sage_gemm_kernel<<<gemm_blocks, TB, 0, stream>>>(hB, hA, inv, W2l, b2l, W2r, hB, nrows);

  // ---- layer 3: scatter hB -> hA, GEMM(hA agg, hB self) -> hA ----
  zero4_kernel<<<blocks(feat / 4, TB), TB, 0, stream>>>((float4*)hA, feat / 4);
  scatter_add_kernel<<<blocks(E * 64, TB), TB, 0, stream>>>(hB, src, dst, hA, E * 64);
  sage_gemm_kernel<<<gemm_blocks, TB, 0, stream>>>(hA, hB, inv, W3l, b3l, W3r, hA, nrows);

  // ---- pool + readout ----
  zero4_kernel<<<blocks((64 * 256 + 64) / 4, TB), TB, 0, stream>>>((float4*)pooled,
                                                                   (64 * 256 + 64) / 4);
  pool_kernel<<<blocks(N * 64, TB), TB, 0, stream>>>(hA, batch, pooled, gcnt, N * 64);
  final_kernel<<<1, 64, 0, stream>>>(pooled, gcnt, Wlin, blin, out, 64);
}